// LocalAggregator_24352464569958
// MI455X (gfx1250) — compile-verified
//
#include <hip/hip_runtime.h>

typedef float v2f __attribute__((ext_vector_type(2)));
typedef float v8f __attribute__((ext_vector_type(8)));

#define NB   8
#define NPT  131072
#define NS   1024
#define NK   64
#define NC   32

static __device__ __forceinline__ v8f wmma4(v2f a, v2f b, v8f c) {
  // D = A(16x4 f32) * B(4x16 f32) + C(16x16 f32), exact fp32 path on CDNA5
  return __builtin_amdgcn_wmma_f32_16x16x4_f32(false, a, false, b, (short)0, c,
                                               false, false);
}

static __device__ __forceinline__ v8f vzero8() {
  v8f z = {0.f, 0.f, 0.f, 0.f, 0.f, 0.f, 0.f, 0.f};
  return z;
}

// ---------------------------------------------------------------------------
// K1: feats [B,C,N] -> feats_nc [B,N,C], tiled transpose through LDS
// ---------------------------------------------------------------------------
__global__ __launch_bounds__(256) void k_transpose(const float* __restrict__ feats,
                                                   float* __restrict__ feats_nc) {
  __shared__ float tile[32][65];
  const int bid = blockIdx.x;
  const int b = bid >> 11;               // N/64 = 2048 tiles per batch
  const int n0 = (bid & 2047) << 6;
  const int t = threadIdx.x;
  for (int e = t; e < 32 * 64; e += 256) {
    int c = e >> 6, n = e & 63;
    tile[c][n] = feats[(b * NC + c) * NPT + n0 + n];   // coalesced over n
  }
  __syncthreads();
  for (int e = t; e < 32 * 64; e += 256) {
    int n = e >> 5, c = e & 31;
    feats_nc[(b * NPT + n0 + n) * NC + c] = tile[c][n]; // coalesced over c
  }
}

// ---------------------------------------------------------------------------
// K2: fps outputs (fps_locs, fps_boxes, fps_inds-as-float) + fps (loc,dim) table
// ---------------------------------------------------------------------------
__global__ __launch_bounds__(256) void k_fps(const float* __restrict__ locs,
                                             const float* __restrict__ boxes,
                                             const int* __restrict__ fps_inds,
                                             float* __restrict__ out,
                                             float* __restrict__ fps_pts) {
  const int i = blockIdx.x * 256 + threadIdx.x;   // 0..8191 = b*S+s
  if (i >= NB * NS) return;
  const int b = i >> 10;
  const int idx = fps_inds[i];
  const float* lp = locs + (size_t)(b * NPT + idx) * 3;
  const float* bp = boxes + (size_t)(b * NPT + idx) * 6;
  float l0 = lp[0], l1 = lp[1], l2 = lp[2];
  float b0 = bp[0], b1 = bp[1], b2 = bp[2], b3 = bp[3], b4 = bp[4], b5 = bp[5];
  out[i * 3 + 0] = l0; out[i * 3 + 1] = l1; out[i * 3 + 2] = l2;   // fps_locs
  float* ob = out + 548864 + i * 6;                                 // fps_boxes
  ob[0] = b0; ob[1] = b1; ob[2] = b2; ob[3] = b3; ob[4] = b4; ob[5] = b5;
  out[598016 + i] = (float)idx;                                     // fps_inds
  float* fp = fps_pts + i * 6;
  fp[0] = l0; fp[1] = l1; fp[2] = l2;
  fp[3] = b3 - b0; fp[4] = b4 - b1; fp[5] = b5 - b2;
}

// ---------------------------------------------------------------------------
// K2b: pack stage-3 weights into B-fragment order [(kk*Ncol+col)*4 + kr]
//      so one global_load_b64 yields a whole 4x16 B-fragment pair
// ---------------------------------------------------------------------------
__global__ __launch_bounds__(256) void k_packw3(const float* __restrict__ w3a,
                                                const float* __restrict__ w3b,
                                                float* __restrict__ w3a_p,
                                                float* __restrict__ w3b_p) {
  const int i = blockIdx.x * 256 + threadIdx.x;    // 0..32767
  if (i < 16384) {                                 // w3a: K=64, Ncol=256
    int kk = i >> 10, col = (i >> 2) & 255, kr = i & 3;
    w3a_p[i] = w3a[(kk * 4 + kr) * 256 + col];
  } else {                                         // w3b: K=256, Ncol=64
    int j = i - 16384;
    int kk = j >> 8, col = (j >> 2) & 63, kr = j & 3;
    w3b_p[j] = w3b[(kk * 4 + kr) * 64 + col];
  }
}

// ---------------------------------------------------------------------------
// K3: stage-1 gather + MLP(38->32 relu ->64 relu) + max over K, via WMMA f32
// one wave per (b,s); 2 waves per block; frag-packed weights in LDS
// ---------------------------------------------------------------------------
__global__ __launch_bounds__(64) void k_stage1(
    const float* __restrict__ locs, const float* __restrict__ boxes,
    const float* __restrict__ feats_nc, const int* __restrict__ nbr,
    const float* __restrict__ fps_pts,
    const float* __restrict__ w1a, const float* __restrict__ s1a,
    const float* __restrict__ t1a, const float* __restrict__ w1b,
    const float* __restrict__ s1b, const float* __restrict__ t1b,
    float* __restrict__ nf_ws) {
  extern __shared__ float sm[];
  float* w1a_p = sm;              // frag-packed, 10 kk x 32 col x 4 = 1280
  float* s1a_s = sm + 1280;
  float* t1a_s = sm + 1312;
  float* w1b_p = sm + 1344;       // frag-packed, 8 kk x 64 col x 4 = 2048
  float* s1b_s = sm + 3392;
  float* t1b_s = sm + 3456;
  const int t = threadIdx.x;
  const int wave = t >> 5, lane = t & 31;
  float* g  = sm + 3520 + wave * (64 * 42);   // gathered 64 x 40 (stride 42)
  float* h1 = sm + 8896 + wave * (64 * 34);   // hidden   64 x 32 (stride 34)

  for (int i = t; i < 1280; i += 64) {        // zero-padded K rows 38,39
    int kk = i >> 7, col = (i >> 2) & 31, kr = i & 3, k = kk * 4 + kr;
    w1a_p[i] = (k < 38) ? w1a[k * 32 + col] : 0.f;
  }
  for (int i = t; i < 2048; i += 64) {
    int kk = i >> 8, col = (i >> 2) & 63, kr = i & 3;
    w1b_p[i] = w1b[(kk * 4 + kr) * 64 + col];
  }
  if (t < 32) { s1a_s[t] = s1a[t]; t1a_s[t] = t1a[t]; }
  if (t < 64) { s1b_s[t] = s1b[t]; t1b_s[t] = t1b[t]; }
  __syncthreads();

  const int unit = blockIdx.x * 2 + wave;
  const int b = unit >> 10, s = unit & 1023;
  const float* ctr = fps_pts + (b * NS + s) * 6;
  const float c0 = ctr[0], c1 = ctr[1], c2 = ctr[2];
  const float c3 = ctr[3], c4 = ctr[4], c5 = ctr[5];

  // gather: each lane fills 2 rows of g
  for (int rr = 0; rr < 2; ++rr) {
    const int r = lane + rr * 32;
    const int gi = nbr[(b * NS + s) * NK + r];
    const float* lp = locs + (size_t)(b * NPT + gi) * 3;
    const float* bp = boxes + (size_t)(b * NPT + gi) * 6;
    float* gr = g + r * 42;
    gr[0] = (lp[0] - c0) * 2.5f;   // 1/0.4
    gr[1] = (lp[1] - c1) * 2.5f;
    gr[2] = (lp[2] - c2) * 2.5f;
    gr[3] = fabsf((bp[3] - bp[0]) - c3);
    gr[4] = fabsf((bp[4] - bp[1]) - c4);
    gr[5] = fabsf((bp[5] - bp[2]) - c5);
    const float4* fp4 = (const float4*)(feats_nc + (size_t)(b * NPT + gi) * NC);
    #pragma unroll
    for (int q = 0; q < 8; ++q) {
      float4 v = fp4[q];
      gr[6 + 4 * q + 0] = v.x; gr[6 + 4 * q + 1] = v.y;
      gr[6 + 4 * q + 2] = v.z; gr[6 + 4 * q + 3] = v.w;
    }
    gr[38] = 0.f; gr[39] = 0.f;    // K padding
  }
  __syncthreads();

  const int lm = lane & 15;
  const int koff = (lane < 16) ? 0 : 2;   // A/B-frag K offset per lane half
  const int mofs = (lane < 16) ? 0 : 8;   // C/D row offset per lane half

  // mlp1a: [64x40] @ [40x32] -> h1 (relu); B-frags cached across M-tiles
  #pragma unroll
  for (int nt = 0; nt < 2; ++nt) {
    v2f Bf[10];
    #pragma unroll
    for (int kk = 0; kk < 10; ++kk)
      Bf[kk] = *(const v2f*)(w1a_p + (kk * 32 + nt * 16 + lm) * 4 + koff);
    const float sc = s1a_s[nt * 16 + lm], sh = t1a_s[nt * 16 + lm];
    for (int mt = 0; mt < 4; ++mt) {
      v8f acc = vzero8();
      #pragma unroll
      for (int kk = 0; kk < 10; ++kk) {
        const v2f a = *(const v2f*)(g + (mt * 16 + lm) * 42 + kk * 4 + koff);
        acc = wmma4(a, Bf[kk], acc);
      }
      #pragma unroll
      for (int r = 0; r < 8; ++r) {
        h1[(mt * 16 + mofs + r) * 34 + nt * 16 + lm] =
            fmaxf(acc[r] * sc + sh, 0.f);
      }
    }
  }
  __syncthreads();

  // mlp1b: [64x32] @ [32x64] -> relu -> column max over 64 rows
  float cps[4] = {0.f, 0.f, 0.f, 0.f};   // relu output >= 0
  #pragma unroll
  for (int nt = 0; nt < 4; ++nt) {
    v2f Bf[8];
    #pragma unroll
    for (int kk = 0; kk < 8; ++kk)
      Bf[kk] = *(const v2f*)(w1b_p + (kk * 64 + nt * 16 + lm) * 4 + koff);
    const float sc = s1b_s[nt * 16 + lm], sh = t1b_s[nt * 16 + lm];
    for (int mt = 0; mt < 4; ++mt) {
      v8f acc = vzero8();
      #pragma unroll
      for (int kk = 0; kk < 8; ++kk) {
        const v2f a = *(const v2f*)(h1 + (mt * 16 + lm) * 34 + kk * 4 + koff);
        acc = wmma4(a, Bf[kk], acc);
      }
      float m = 0.f;
      #pragma unroll
      for (int r = 0; r < 8; ++r) m = fmaxf(m, fmaxf(acc[r] * sc + sh, 0.f));
      cps[nt] = fmaxf(cps[nt], m);
    }
  }
  #pragma unroll
  for (int nt = 0; nt < 4; ++nt) {
    const float o = __shfl_xor(cps[nt], 16, 32);
    const float mx = fmaxf(cps[nt], o);
    if (lane < 16) nf_ws[(b * NS + s) * 64 + nt * 16 + lane] = mx;
  }
}

// ---------------------------------------------------------------------------
// K4: stage-2 gather + MLP(70->64, no relu) + max over K2
// ---------------------------------------------------------------------------
__global__ __launch_bounds__(64) void k_stage2(
    const float* __restrict__ fps_pts, const float* __restrict__ nf_ws,
    const int* __restrict__ nbr2, const float* __restrict__ w2,
    const float* __restrict__ s2, const float* __restrict__ t2,
    float* __restrict__ h2_ws) {
  extern __shared__ float sm[];
  float* w2_p = sm;               // frag-packed, 18 kk x 64 col x 4 = 4608
  float* s2_s = sm + 4608;
  float* t2_s = sm + 4672;
  const int t = threadIdx.x;
  const int wave = t >> 5, lane = t & 31;
  float* g2 = sm + 4736 + wave * (64 * 74);   // 64 x 72 (stride 74)

  for (int i = t; i < 4608; i += 64) {        // zero-padded K rows 70,71
    int kk = i >> 8, col = (i >> 2) & 63, kr = i & 3, k = kk * 4 + kr;
    w2_p[i] = (k < 70) ? w2[k * 64 + col] : 0.f;
  }
  if (t < 64) { s2_s[t] = s2[t]; t2_s[t] = t2[t]; }
  __syncthreads();

  const int unit = blockIdx.x * 2 + wave;
  const int b = unit >> 10, s = unit & 1023;
  const float* ctr = fps_pts + (b * NS + s) * 6;
  const float c0 = ctr[0], c1 = ctr[1], c2 = ctr[2];
  const float c3 = ctr[3], c4 = ctr[4], c5 = ctr[5];

  for (int rr = 0; rr < 2; ++rr) {
    const int r = lane + rr * 32;
    const int gi = nbr2[(b * NS + s) * NK + r];    // index into [0,S)
    const float* sp = fps_pts + (b * NS + gi) * 6;
    float* gr = g2 + r * 74;
    gr[0] = (sp[0] - c0) * 1.25f;   // 1/0.8
    gr[1] = (sp[1] - c1) * 1.25f;
    gr[2] = (sp[2] - c2) * 1.25f;
    gr[3] = fabsf(sp[3] - c3);
    gr[4] = fabsf(sp[4] - c4);
    gr[5] = fabsf(sp[5] - c5);
    const float4* nf4 = (const float4*)(nf_ws + (size_t)(b * NS + gi) * 64);
    #pragma unroll
    for (int q = 0; q < 16; ++q) {
      float4 v = nf4[q];
      gr[6 + 4 * q + 0] = v.x; gr[6 + 4 * q + 1] = v.y;
      gr[6 + 4 * q + 2] = v.z; gr[6 + 4 * q + 3] = v.w;
    }
    gr[70] = 0.f; gr[71] = 0.f;
  }
  __syncthreads();

  const int lm = lane & 15;
  const int koff = (lane < 16) ? 0 : 2;
  const float NINF = -__builtin_inff();

  float cps[4] = {NINF, NINF, NINF, NINF};
  #pragma unroll
  for (int nt = 0; nt < 4; ++nt) {
    v2f Bf[18];
    #pragma unroll
    for (int kk = 0; kk < 18; ++kk)
      Bf[kk] = *(const v2f*)(w2_p + (kk * 64 + nt * 16 + lm) * 4 + koff);
    const float sc = s2_s[nt * 16 + lm], sh = t2_s[nt * 16 + lm];
    for (int mt = 0; mt < 4; ++mt) {
      v8f acc = vzero8();
      #pragma unroll
      for (int kk = 0; kk < 18; ++kk) {
        const v2f a = *(const v2f*)(g2 + (mt * 16 + lm) * 74 + kk * 4 + koff);
        acc = wmma4(a, Bf[kk], acc);
      }
      float m = NINF;
      #pragma unroll
      for (int r = 0; r < 8; ++r) m = fmaxf(m, acc[r] * sc + sh);  // no relu
      cps[nt] = fmaxf(cps[nt], m);
    }
  }
  #pragma unroll
  for (int nt = 0; nt < 4; ++nt) {
    const float o = __shfl_xor(cps[nt], 16, 32);
    const float mx = fmaxf(cps[nt], o);
    if (lane < 16) h2_ws[(b * NS + s) * 64 + nt * 16 + lane] = mx;
  }
}

// ---------------------------------------------------------------------------
// K5: stage-3 bottleneck GEMM 64->256(relu)->64 + skip + relu, transposed store
// one wave per 16-row tile; A-frags of gemm1 cached in regs across all N-tiles
// ---------------------------------------------------------------------------
__global__ __launch_bounds__(64) void k_stage3(
    const float* __restrict__ h2_ws, const float* __restrict__ nf_ws,
    const float* __restrict__ w3a_p, const float* __restrict__ s3a,
    const float* __restrict__ t3a, const float* __restrict__ w3b_p,
    const float* __restrict__ s3b, const float* __restrict__ t3b,
    float* __restrict__ out_ch) {
  extern __shared__ float sm[];
  const int t = threadIdx.x;
  const int wave = t >> 5, lane = t & 31;
  float* h3 = sm + wave * (16 * 274);          // 16 x 256 (stride 274: 18 mod 64)

  const int tile = blockIdx.x * 2 + wave;      // 0..511
  const int row0 = tile * 16;
  const int lm = lane & 15;
  const int koff = (lane < 16) ? 0 : 2;
  const int mofs = (lane < 16) ? 0 : 8;

  // gemm1: [16x64] @ [64x256] -> relu -> h3 (LDS); A depends only on kk
  v2f Af[16];
  #pragma unroll
  for (int kk = 0; kk < 16; ++kk)
    Af[kk] = *(const v2f*)(h2_ws + (size_t)(row0 + lm) * 64 + kk * 4 + koff);
  for (int nt = 0; nt < 16; ++nt) {
    v8f acc = vzero8();
    #pragma unroll
    for (int kk = 0; kk < 16; ++kk) {
      const v2f bv =
          *(const v2f*)(w3a_p + (kk * 256 + nt * 16 + lm) * 4 + koff);
      acc = wmma4(Af[kk], bv, acc);
    }
    const float sc = s3a[nt * 16 + lm], sh = t3a[nt * 16 + lm];
    #pragma unroll
    for (int r = 0; r < 8; ++r) {
      h3[(mofs + r) * 274 + nt * 16 + lm] = fmaxf(acc[r] * sc + sh, 0.f);
    }
  }
  __syncthreads();

  // gemm2: [16x256] @ [256x64] + scale/shift + identity + relu, transposed out
  for (int nt = 0; nt < 4; ++nt) {
    v8f acc = vzero8();
    #pragma unroll
    for (int kk = 0; kk < 64; ++kk) {
      const v2f a = *(const v2f*)(h3 + lm * 274 + kk * 4 + koff);
      const v2f bv =
          *(const v2f*)(w3b_p + (kk * 64 + nt * 16 + lm) * 4 + koff);
      acc = wmma4(a, bv, acc);
    }
    const int col = nt * 16 + lm;
    const float sc = s3b[col], sh = t3b[col];
    #pragma unroll
    for (int r = 0; r < 8; ++r) {
      const int row = row0 + mofs + r;         // global (b*S+s)
      const float idv = nf_ws[(size_t)row * 64 + col];
      const float y = fmaxf(acc[r] * sc + sh + idv, 0.f);
      const int bb = row >> 10, ss = row & 1023;
      out_ch[(size_t)bb * 64 * NS + (size_t)col * NS + ss] = y;
    }
  }
}

// ---------------------------------------------------------------------------
extern "C" void kernel_launch(void* const* d_in, const int* in_sizes, int n_in,
                              void* d_out, int out_size, void* d_ws,
                              size_t ws_size, hipStream_t stream) {
  const float* locs  = (const float*)d_in[0];
  const float* feats = (const float*)d_in[1];
  const float* boxes = (const float*)d_in[2];
  const int* fps_inds = (const int*)d_in[3];
  const int* nbr  = (const int*)d_in[4];
  const int* nbr2 = (const int*)d_in[5];
  const float* w1a = (const float*)d_in[6];
  const float* s1a = (const float*)d_in[7];
  const float* t1a = (const float*)d_in[8];
  const float* w1b = (const float*)d_in[9];
  const float* s1b = (const float*)d_in[10];
  const float* t1b = (const float*)d_in[11];
  const float* w2  = (const float*)d_in[12];
  const float* s2  = (const float*)d_in[13];
  const float* t2  = (const float*)d_in[14];
  const float* w3a = (const float*)d_in[15];
  const float* s3a = (const float*)d_in[16];
  const float* t3a = (const float*)d_in[17];
  const float* w3b = (const float*)d_in[18];
  const float* s3b = (const float*)d_in[19];
  const float* t3b = (const float*)d_in[20];

  float* out = (float*)d_out;
  float* ws = (float*)d_ws;
  float* feats_nc = ws;                 // B*N*C       = 33,554,432 floats
  float* nf_ws    = ws + 33554432;      // B*S*64      =    524,288
  float* h2_ws    = ws + 34078720;      // B*S*64      =    524,288
  float* fps_pts  = ws + 34603008;      // B*S*6       =     49,152
  float* w3a_p    = ws + 34652160;      // 64*256      =     16,384 (packed)
  float* w3b_p    = ws + 34668544;      // 256*64      =     16,384 (packed)

  k_transpose<<<NB * (NPT / 64), 256, 0, stream>>>(feats, feats_nc);
  k_fps<<<(NB * NS + 255) / 256, 256, 0, stream>>>(locs, boxes, fps_inds, out,
                                                   fps_pts);
  k_packw3<<<128, 256, 0, stream>>>(w3a, w3b, w3a_p, w3b_p);
  k_stage1<<<NB * NS / 2, 64, 13248 * 4, stream>>>(
      locs, boxes, feats_nc, nbr, fps_pts, w1a, s1a, t1a, w1b, s1b, t1b, nf_ws);
  k_stage2<<<NB * NS / 2, 64, 14208 * 4, stream>>>(fps_pts, nf_ws, nbr2, w2, s2,
                                                   t2, h2_ws);
  k_stage3<<<NB * NS / 16 / 2, 64, 8768 * 4, stream>>>(
      h2_ws, nf_ws, w3a_p, s3a, t3a, w3b_p, s3b, t3b, out + 24576);
}